// Loss_13709535608994
// MI455X (gfx1250) — compile-verified
//
#include <hip/hip_runtime.h>
#include <hip/hip_bf16.h>
#include <math.h>
#include <stdint.h>

#define CC     0.01f
#define SQRTC  0.1f
#define CLIPR  2.3f
#define TAUG   0.1f
#define MRGF   0.1f
#define NB     128
#define NBT    10
#define DIM    512
#define BSZ    16384
#define EPSF   1e-5f
#define CAPP   (BSZ / NB)   // 128
#define TPER   3

// K-chunking for the LDS-staged GEMM
#define KCH    128
#define NCHUNK (DIM / KCH)        // 4
#define ROWPAD (KCH + 1)          // TDM pad: +1 dword per 128-dword row (bank-conflict-free)
#define BUFSZ  (NB * ROWPAD)      // floats per LDS buffer (16512)

typedef __attribute__((ext_vector_type(2))) float v2f;
typedef __attribute__((ext_vector_type(8))) float v8f;
typedef __attribute__((ext_vector_type(4))) unsigned int u32x4;
typedef __attribute__((ext_vector_type(8))) int i32x8;
typedef __attribute__((ext_vector_type(4))) int i32x4;

#if defined(__has_builtin)
#if __has_builtin(__builtin_amdgcn_wmma_f32_16x16x4_f32)
#define HAVE_WMMA_F32X4 1
#endif
#if __has_builtin(__builtin_amdgcn_tensor_load_to_lds)
#define HAVE_TDM 1
#endif
#endif
#ifndef HAVE_WMMA_F32X4
#define HAVE_WMMA_F32X4 0
#endif
#ifndef HAVE_TDM
#define HAVE_TDM 0
#endif

// ---------------- utilities ----------------

__device__ inline float wred_sum(float v) {
  for (int m = 16; m > 0; m >>= 1) v += __shfl_xor(v, m, 32);
  return v;
}
__device__ inline float wred_max(float v) {
  for (int m = 16; m > 0; m >>= 1) v = fmaxf(v, __shfl_xor(v, m, 32));
  return v;
}
__device__ inline void wred_argmax(float& v, int& i) {
  for (int m = 16; m > 0; m >>= 1) {
    float ov = __shfl_xor(v, m, 32);
    int   oi = __shfl_xor(i, m, 32);
    if (ov > v || (ov == v && oi >= 0 && (i < 0 || oi < i))) { v = ov; i = oi; }
  }
}
__device__ inline int wred_min_i(int v) {
  for (int m = 16; m > 0; m >>= 1) v = min(v, __shfl_xor(v, m, 32));
  return v;
}

__device__ inline uint32_t hash3(uint32_t a, uint32_t b, uint32_t c) {
  uint64_t x = (uint64_t)a * 0x9E3779B97F4A7C15ULL;
  x ^= ((uint64_t)b << 32) | c;
  x ^= x >> 30; x *= 0xBF58476D1CE4E5B9ULL;
  x ^= x >> 27; x *= 0x94D049BB133111EBULL;
  x ^= x >> 31;
  return (uint32_t)x;
}
__device__ inline float gumbel_from(uint32_t h) {
  float u = fmaxf((float)(h >> 8) * (1.0f / 16777216.0f), 1e-10f);
  return -logf(-logf(u));
}

// Poincare distance from inner product + squared norms (matches reference dist_matrix)
__device__ inline float poin_dist(float xy, float x2, float y2) {
  float A  = 1.0f - 2.0f * CC * xy + CC * y2;
  float Bv = 1.0f - CC * x2;
  float Dn = 1.0f - 2.0f * CC * xy + CC * CC * x2 * y2;
  float den = fmaxf(Dn * Dn, 1e-15f);
  float sq  = (A * A * x2 - 2.0f * A * Bv * xy + Bv * Bv * y2) / den;
  float nrm = sqrtf(fmaxf(sq, 0.0f));
  float t   = fminf(fmaxf(SQRTC * nrm, 0.0f), 1.0f - EPSF);
  return (2.0f / SQRTC) * atanhf(t);
}

// Issue a TDM load of a 128-row x 128-col f32 tile of lcas_h into LDS at lds_byte_off.
// D# per CDNA5 ISA ch.8: data_size=4B, pad 1 dword per 128 dwords (ROWPAD layout).
__device__ inline void tdm_issue(const float* gsrc, unsigned lds_byte_off) {
#if HAVE_TDM
  unsigned long long ga = (unsigned long long)(uintptr_t)gsrc;
  u32x4 g0 = {};
  g0[0] = 1u;                                               // count=1 (valid user D#)
  g0[1] = lds_byte_off;                                     // lds_addr
  g0[2] = (unsigned)(ga & 0xFFFFFFFFu);                     // global_addr[31:0]
  g0[3] = (unsigned)((ga >> 32) & 0x01FFFFFFu) | (2u << 30);// global_addr[56:32] | type=2
  i32x8 g1 = {};
  g1[0] = (2 << 16)        // data_size = 4 bytes
        | (1 << 20)        // pad_enable
        | (6 << 22)        // pad_interval code 6 -> every 128 dwords
        | (0 << 25);       // pad_amount code 0 -> 1 dword
  g1[1] = (DIM & 0xFFFF) << 16;   // tensor_dim0[15:0] = 512
  g1[2] = (NB  & 0xFFFF) << 16;   // tensor_dim1[15:0] = 128
  g1[3] = (KCH & 0xFFFF) << 16;   // tile_dim0 = 128
  g1[4] = NB;                     // tile_dim1 = 128, tile_dim2 = 0
  g1[5] = DIM;                    // tensor_dim0_stride[31:0] = 512
  g1[6] = 0;
  g1[7] = 0;
  i32x4 z4 = {};
#if defined(__clang_major__) && (__clang_major__ >= 23)
  i32x8 z8 = {};
  __builtin_amdgcn_tensor_load_to_lds(g0, g1, z4, z4, z8, 0);
#else
  __builtin_amdgcn_tensor_load_to_lds(g0, g1, z4, z4, 0);
#endif
#endif
}

// ---------------- kernels ----------------

// zero small scratch (zero vector, zero-x2, accumulators)
__global__ void k_init(float* zerov, float* zx2, float* acc) {
  int t = threadIdx.x;
  for (int i = t; i < DIM; i += blockDim.x) zerov[i] = 0.0f;
  if (t == 0) zx2[0] = 0.0f;
  if (t < 16) acc[t] = 0.0f;
}

// to_poincare: clip -> expmap0 -> project (wave per row)
__global__ void k_poincare(const float* __restrict__ x, float* __restrict__ out, int rows) {
  int gw = (blockIdx.x * blockDim.x + threadIdx.x) >> 5;
  int lane = threadIdx.x & 31;
  if (gw >= rows) return;
  const float* xr = x + (size_t)gw * DIM;
  float xv[DIM / 32];
  float s = 0.0f;
#pragma unroll
  for (int t = 0; t < DIM / 32; ++t) { xv[t] = xr[lane + t * 32]; s += xv[t] * xv[t]; }
  s = wred_sum(s);
  float n  = fmaxf(sqrtf(s), 1e-15f);
  float sc = fminf(1.0f, CLIPR / n);
  float xn = fmaxf(n * sc, 1e-15f);
  float f  = tanhf(SQRTC * xn) / (SQRTC * xn);
  float pn = fmaxf(f * xn, 1e-15f);
  float maxnorm = (1.0f - 1e-3f) / SQRTC;
  float fix = (pn > maxnorm) ? (maxnorm / pn) : 1.0f;
  float tot = sc * f * fix;
  float* orow = out + (size_t)gw * DIM;
#pragma unroll
  for (int t = 0; t < DIM / 32; ++t) orow[lane + t * 32] = xv[t] * tot;
}

// squared row norms (wave per row)
__global__ void k_rownorm2(const float* __restrict__ x, float* __restrict__ out, int rows) {
  int gw = (blockIdx.x * blockDim.x + threadIdx.x) >> 5;
  int lane = threadIdx.x & 31;
  if (gw >= rows) return;
  const float* xr = x + (size_t)gw * DIM;
  float s = 0.0f;
#pragma unroll
  for (int t = 0; t < DIM / 32; ++t) { float v = xr[lane + t * 32]; s += v * v; }
  s = wred_sum(s);
  if (lane == 0) out[gw] = s;
}

// Big fused GEMM: sim = -poincare_dist(z, lcas_h) for z_s and z_s_1 simultaneously.
// One wave = one 16x16 tile; 8 waves/block cover all NB=128 columns for 16 rows.
// B (lcas_h) is staged through LDS by the Tensor Data Mover, double-buffered over
// K-chunks of 128, with TDM row padding (stride 129 dwords) to kill bank conflicts.
__global__ void k_gemm_sim(const float* __restrict__ zs, const float* __restrict__ zs1,
                           const float* __restrict__ lh,
                           const float* __restrict__ x2, const float* __restrict__ x2_1,
                           const float* __restrict__ y2,
                           float* __restrict__ sim, float* __restrict__ sim1) {
  extern __shared__ float bsh[];   // 2 * BUFSZ floats
  int wave = threadIdx.x >> 5;
  int lane = threadIdx.x & 31;
  int half = lane >> 4;            // 0: K+0..1, 1: K+2..3
  int l16  = lane & 15;
  int tileM = blockIdx.x * 16;
  int n0    = wave * 16;

  const float* Arow  = zs  + (size_t)(tileM + l16) * DIM;
  const float* Arow1 = zs1 + (size_t)(tileM + l16) * DIM;

  // stage chunk 0 into buffer 0
#if HAVE_TDM
  if (wave == 0) {
    tdm_issue(lh, 0u);
    __builtin_amdgcn_s_wait_tensorcnt(0);
  }
#else
  for (int idx = threadIdx.x; idx < NB * KCH; idx += blockDim.x) {
    int r = idx / KCH, kk = idx - r * KCH;
    bsh[r * ROWPAD + kk] = lh[(size_t)r * DIM + kk];
  }
#endif
  __syncthreads();

  v8f acc0 = {};
  v8f acc1 = {};

  for (int c = 0; c < NCHUNK; ++c) {
    int buf = c & 1;
    int k0  = c * KCH;

    // issue next chunk (overlaps with WMMA below on the TDM path)
    if (c + 1 < NCHUNK) {
#if HAVE_TDM
      if (wave == 0)
        tdm_issue(lh + (size_t)(k0 + KCH), (unsigned)(((c + 1) & 1) * BUFSZ * 4));
#else
      int nb = ((c + 1) & 1) * BUFSZ;
      for (int idx = threadIdx.x; idx < NB * KCH; idx += blockDim.x) {
        int r = idx / KCH, kk = idx - r * KCH;
        bsh[nb + r * ROWPAD + kk] = lh[(size_t)r * DIM + (k0 + KCH) + kk];
      }
#endif
    }

    __builtin_prefetch(Arow + k0 + KCH, 0, 3);
    __builtin_prefetch(Arow1 + k0 + KCH, 0, 3);

    const float* Brow = bsh + buf * BUFSZ + (n0 + l16) * ROWPAD;
    for (int kk = 0; kk < KCH; kk += 4) {
      int ka = k0 + kk + half * 2;
      int kb = kk + half * 2;
      v2f a, a1, b;
      a.x  = Arow[ka];   a.y  = Arow[ka + 1];
      a1.x = Arow1[ka];  a1.y = Arow1[ka + 1];
      b.x  = Brow[kb];   b.y  = Brow[kb + 1];
#if HAVE_WMMA_F32X4
      acc0 = __builtin_amdgcn_wmma_f32_16x16x4_f32(false, a, false, b, (short)0, acc0, false, false);
      acc1 = __builtin_amdgcn_wmma_f32_16x16x4_f32(false, a1, false, b, (short)0, acc1, false, false);
#else
      acc0[0] += a.x * b.x + a.y * b.y;   // placeholder if builtin absent
      acc1[0] += a1.x * b.x + a1.y * b.y;
#endif
    }

#if HAVE_TDM
    if (wave == 0 && c + 1 < NCHUNK) __builtin_amdgcn_s_wait_tensorcnt(0);
#endif
    __syncthreads();
  }

#pragma unroll
  for (int r = 0; r < 8; ++r) {
    int m = tileM + r + half * 8;
    int n = n0 + l16;
    float y2n = y2[n];
    sim [(size_t)m * NB + n] = -poin_dist(acc0[r], x2[m],   y2n);
    sim1[(size_t)m * NB + n] = -poin_dist(acc1[r], x2_1[m], y2n);
  }
}

// Sequential greedy capacity-constrained assignment (single wave, faithful to scan):
// pick argmax of class-masked sim among prototypes with count<cap; fallback = lowest
// index with spare capacity (matches stable argsort of -inf tail).
__global__ void k_assign(const float* __restrict__ sim, const int* __restrict__ y,
                         int* __restrict__ select) {
  __shared__ int counts[NB];
  int lane = threadIdx.x;
#pragma unroll
  for (int t = 0; t < NB / 32; ++t) counts[lane + t * 32] = 0;
  __syncthreads();
  for (int row = 0; row < BSZ; ++row) {
    int yy = y[row];
    float bv = -INFINITY; int bi = -1;
    int fb = 0x7FFFFFFF;
#pragma unroll
    for (int t = 0; t < NB / 32; ++t) {
      int p = lane + t * 32;
      bool dis   = (yy == 0) ? (p >= NB / 2) : (p < NB / 2);
      bool avail = counts[p] < CAPP;
      if (avail) fb = min(fb, p);
      if (avail && !dis) {
        float v = sim[(size_t)row * NB + p];
        if (v > bv || (v == bv && p < bi)) { bv = v; bi = p; }
      }
    }
    wred_argmax(bv, bi);
    fb = wred_min_i(fb);
    int sel = (bi >= 0) ? bi : ((fb != 0x7FFFFFFF) ? fb : 0);
    if (lane == 0) {
      if (fb != 0x7FFFFFFF) counts[sel] += 1;
      select[row] = sel;
    }
    __syncthreads();
  }
}

// Per-row losses: CE vs select_prot (sim & sim_1), linear-head CE vs y, and
// sum((sim - sim_1)^2). Wave per row; f32 atomics into accumulators.
__global__ void k_row_losses(const float* __restrict__ sim, const float* __restrict__ sim1,
                             const int* __restrict__ select, const int* __restrict__ y,
                             const float* __restrict__ W, float* __restrict__ acc) {
  int gw = (blockIdx.x * blockDim.x + threadIdx.x) >> 5;
  int lane = threadIdx.x & 31;
  if (gw >= BSZ) return;
  const float* sr  = sim  + (size_t)gw * NB;
  const float* sr1 = sim1 + (size_t)gw * NB;
  float v[NB / 32], v1[NB / 32];
  float mx = -INFINITY, mx1 = -INFINITY, dd = 0.0f;
  float c0 = 0.0f, c1 = 0.0f, c0b = 0.0f, c1b = 0.0f;
#pragma unroll
  for (int t = 0; t < NB / 32; ++t) {
    int p = lane + t * 32;
    v[t] = sr[p]; v1[t] = sr1[p];
    mx = fmaxf(mx, v[t]); mx1 = fmaxf(mx1, v1[t]);
    float d = v[t] - v1[t]; dd += d * d;
    c0  += v[t]  * W[p];       c1  += v[t]  * W[NB + p];
    c0b += v1[t] * W[p];       c1b += v1[t] * W[NB + p];
  }
  mx = wred_max(mx); mx1 = wred_max(mx1);
  float se = 0.0f, se1 = 0.0f;
#pragma unroll
  for (int t = 0; t < NB / 32; ++t) { se += expf(v[t] - mx); se1 += expf(v1[t] - mx1); }
  se = wred_sum(se); se1 = wred_sum(se1);
  dd = wred_sum(dd);
  c0 = wred_sum(c0); c1 = wred_sum(c1); c0b = wred_sum(c0b); c1b = wred_sum(c1b);
  if (lane == 0) {
    int lab = select[gw];
    float lse  = mx + logf(se);
    float lse1 = mx1 + logf(se1);
    float cePI  = lse  - sr[lab];
    float cePI1 = lse1 - sr1[lab];
    int yy = y[gw];
    float m2 = fmaxf(c0, c1);
    float l2 = m2 + logf(expf(c0 - m2) + expf(c1 - m2));
    float ceC = l2 - (yy == 0 ? c0 : c1);
    float m2b = fmaxf(c0b, c1b);
    float l2b = m2b + logf(expf(c0b - m2b) + expf(c1b - m2b));
    float ceC1 = l2b - (yy == 0 ? c0b : c1b);
    atomicAdd(&acc[0], cePI);
    atomicAdd(&acc[1], cePI1);
    atomicAdd(&acc[2], ceC);
    atomicAdd(&acc[3], ceC1);
    atomicAdd(&acc[4], dd);
  }
}

// sum((z_s - z_s_1)^2) over BSZ*DIM (grid-stride, wave-level atomic)
__global__ void k_zdiff(const float* __restrict__ a, const float* __restrict__ b,
                        float* __restrict__ acc) {
  size_t n = (size_t)BSZ * DIM;
  size_t stride = (size_t)gridDim.x * blockDim.x;
  float s = 0.0f;
  for (size_t i = blockIdx.x * (size_t)blockDim.x + threadIdx.x; i < n; i += stride) {
    float d = a[i] - b[i];
    s += d * d;
  }
  s = wred_sum(s);
  if ((threadIdx.x & 31) == 0) atomicAdd(&acc[5], s);
}

// small Poincare distance matrix: D[i,j] (thread per element, 512-dot inline)
__global__ void k_dist_small(const float* __restrict__ X, const float* __restrict__ x2v, int nx,
                             const float* __restrict__ Y, const float* __restrict__ y2v, int ny,
                             float* __restrict__ D) {
  int idx = blockIdx.x * blockDim.x + threadIdx.x;
  if (idx >= nx * ny) return;
  int i = idx / ny, j = idx % ny;
  const float* xr = X + (size_t)i * DIM;
  const float* yr = Y + (size_t)j * DIM;
  float dot = 0.0f;
  for (int k = 0; k < DIM; ++k) dot += xr[k] * yr[k];
  D[idx] = poin_dist(dot, x2v[i], y2v[j]);
}

// accumulation of log1p(sum_j exp(-D[row,j])) (wave per row)
__global__ void k_proto_lse(const float* __restrict__ D, int cols, float* __restrict__ accSlot) {
  int row = blockIdx.x;
  int lane = threadIdx.x;
  float s = 0.0f;
  for (int j = lane; j < cols; j += 32) s += expf(-D[(size_t)row * cols + j]);
  s = wred_sum(s);
  if (lane == 0) atomicAdd(accSlot, log1pf(s));
}

// sim_leaf = exp(-D_ll) + same-half-label bonus ; sim_top = exp(-D_tt)
__global__ void k_build_simleaf(const float* __restrict__ Dll, float* __restrict__ SL) {
  int idx = blockIdx.x * blockDim.x + threadIdx.x;
  if (idx >= NB * NB) return;
  int i = idx / NB, j = idx % NB;
  float v = expf(-Dll[idx]);
  if ((i >= NB / 2) == (j >= NB / 2)) v += 1.0f;
  SL[idx] = v;
}
__global__ void k_build_simtop(const float* __restrict__ Dtt, float* __restrict__ ST) {
  int idx = blockIdx.x * blockDim.x + threadIdx.x;
  if (idx >= NBT * NBT) return;
  ST[idx] = expf(-Dtt[idx]);
}

// top-3 per row -> one-hot nn matrix (wave per row; stable tie-break = lowest index)
__global__ void k_topk_nn(const float* __restrict__ S, int n, float* __restrict__ NN) {
  int row = blockIdx.x;
  int lane = threadIdx.x;
  const float* sr = S + (size_t)row * n;
  for (int j = lane; j < n; j += 32) NN[(size_t)row * n + j] = 0.0f;
  int picked[3] = {-1, -1, -1};
  for (int t = 0; t < 3; ++t) {
    float bv = -INFINITY; int bi = -1;
    for (int j = lane; j < n; j += 32) {
      if (j == picked[0] || j == picked[1] || j == picked[2]) continue;
      float val = sr[j];
      if (val > bv || (val == bv && (bi < 0 || j < bi))) { bv = val; bi = j; }
    }
    wred_argmax(bv, bi);
    picked[t] = bi;
    if (lane == 0 && bi >= 0) NN[(size_t)row * n + bi] = 1.0f;
  }
}

// reciprocal triplets: mutual-topk positives; deterministic hash-based sampling
__global__ void k_triplets(const float* __restrict__ NN, int n,
                           int* __restrict__ anc, int* __restrict__ pos, int* __restrict__ neg,
                           float* __restrict__ valid, uint32_t seed) {
  int i = blockIdx.x * blockDim.x + threadIdx.x;
  if (i >= n) return;
  int npos = 0, nneg = 0;
  for (int j = 0; j < n; ++j) {
    float s = 0.5f * (NN[(size_t)i * n + j] + NN[(size_t)j * n + i]);
    if (j == i) s = -1.0f;
    if (s == 1.0f) npos++; else nneg++;
  }
  valid[i] = (npos >= 2) ? 1.0f : 0.0f;
  for (int t = 0; t < TPER; ++t) {
    uint32_t rp = hash3(seed, (uint32_t)i, (uint32_t)(2 * t));
    uint32_t rn = hash3(seed, (uint32_t)i, (uint32_t)(2 * t + 1));
    int pi = 0;
    if (npos > 0) {
      int k = (int)(rp % (uint32_t)npos), c = 0;
      for (int j = 0; j < n; ++j) {
        float s = 0.5f * (NN[(size_t)i * n + j] + NN[(size_t)j * n + i]);
        if (j == i) s = -1.0f;
        if (s == 1.0f) { if (c == k) { pi = j; break; } c++; }
      }
    } else {
      pi = (int)(rp % (uint32_t)n);   // reference: empty pos row -> uniform over all
    }
    int ni = 0;
    {
      int k = (int)(rn % (uint32_t)nneg), c = 0;
      for (int j = 0; j < n; ++j) {
        float s = 0.5f * (NN[(size_t)i * n + j] + NN[(size_t)j * n + i]);
        if (j == i) s = -1.0f;
        if (s < 1.0f) { if (c == k) { ni = j; break; } c++; }
      }
    }
    anc[i * TPER + t] = i;
    pos[i * TPER + t] = pi;
    neg[i * TPER + t] = ni;
  }
}

// GHHC term with gumbel-hard (straight-through forward == hard one-hot pick)
__global__ void k_ghhc(const float* __restrict__ D, const int* __restrict__ anc,
                       const int* __restrict__ pos, const int* __restrict__ neg,
                       const float* __restrict__ valid, int ntrip, uint32_t seed,
                       float* __restrict__ accHC, float* __restrict__ accV) {
  int t = blockIdx.x * blockDim.x + threadIdx.x;
  if (t >= ntrip) return;
  int i = anc[t], j = pos[t], k = neg[t];
  float vi = valid[i];
  const float* di = D + (size_t)i * NBT;
  const float* dj = D + (size_t)j * NBT;
  const float* dk = D + (size_t)k * NBT;
  float l1m = -INFINITY, l2m = -INFINITY;
  int a = 0, b = 0;
#pragma unroll
  for (int c = 0; c < NBT; ++c) {
    float mij  = fmaxf(di[c], dj[c]);
    float mijk = fmaxf(dk[c], mij);
    float l1 = -mij  / TAUG + gumbel_from(hash3(seed,           (uint32_t)t, (uint32_t)(2 * c)));
    float l2 = -mijk / TAUG + gumbel_from(hash3(seed ^ 0x9E37u, (uint32_t)t, (uint32_t)(2 * c + 1)));
    if (l1 > l1m) { l1m = l1; a = c; }
    if (l2 > l2m) { l2m = l2; b = c; }
  }
  float hc = fmaxf(di[a] - di[b] + MRGF, 0.0f)
           + fmaxf(dj[a] - dj[b] + MRGF, 0.0f)
           + fmaxf(dk[b] - dk[a] + MRGF, 0.0f);
  hc *= (a != b ? 1.0f : 0.0f) * vi;
  atomicAdd(accHC, hc);
  atomicAdd(accV, vi);
}

__global__ void k_finalize(const float* __restrict__ acc, float* __restrict__ out) {
  if (threadIdx.x != 0 || blockIdx.x != 0) return;
  float loss_PI   = acc[0] / (float)BSZ;
  float loss_PI1  = acc[1] / (float)BSZ;
  float loss_pred = 0.5f * (acc[2] + acc[3]) / (float)BSZ;
  float loss_II   = acc[4] / ((float)BSZ * (float)NB) + acc[5] / ((float)BSZ * (float)DIM);
  float ghhc_leaf = acc[6] / fmaxf(acc[7], 1.0f);
  float ghhc_top  = acc[8] / fmaxf(acc[9], 1.0f);
  float lse_proto = acc[10] / (float)NB;
  float lse_orig  = acc[11];
  out[0] = loss_PI + loss_PI1 + loss_II
         + (ghhc_leaf + ghhc_top + lse_proto + lse_orig)
         + loss_pred;
}

// ---------------- launch ----------------

extern "C" void kernel_launch(void* const* d_in, const int* in_sizes, int n_in,
                              void* d_out, int out_size, void* d_ws, size_t ws_size,
                              hipStream_t stream) {
  const float* z_s      = (const float*)d_in[0];
  const float* z_s_1    = (const float*)d_in[1];
  const int*   y        = (const int*)d_in[2];
  const float* lcas     = (const float*)d_in[3];
  const float* lcas_top = (const float*)d_in[4];
  const float* linear_w = (const float*)d_in[5];
  float* out = (float*)d_out;

  float* ws = (float*)d_ws;
  size_t off = 0;
  auto alloc = [&](size_t nf) { float* p = ws + off; off += (nf + 63) & ~(size_t)63; return p; };

  float* lcas_h = alloc((size_t)NB * DIM);
  float* ltop_h = alloc((size_t)NBT * DIM);
  float* zerov  = alloc(DIM);
  float* zx2    = alloc(1);
  float* y2     = alloc(NB);
  float* y2t    = alloc(NBT);
  float* x2     = alloc(BSZ);
  float* x2_1   = alloc(BSZ);
  float* sim    = alloc((size_t)BSZ * NB);
  float* sim1   = alloc((size_t)BSZ * NB);
  float* Dll    = alloc((size_t)NB * NB);
  float* Dlt    = alloc((size_t)NB * NBT);
  float* Dtt    = alloc((size_t)NBT * NBT);
  float* Dor    = alloc(NB);
  float* SL     = alloc((size_t)NB * NB);
  float* ST     = alloc((size_t)NBT * NBT);
  float* NNl    = alloc((size_t)NB * NB);
  float* NNt    = alloc((size_t)NBT * NBT);
  int*   selp   = (int*)alloc(BSZ);
  int*   ancl   = (int*)alloc(NB * TPER);
  int*   posl   = (int*)alloc(NB * TPER);
  int*   negl   = (int*)alloc(NB * TPER);
  float* vall   = alloc(NB);
  int*   anct   = (int*)alloc(NBT * TPER);
  int*   post   = (int*)alloc(NBT * TPER);
  int*   negt   = (int*)alloc(NBT * TPER);
  float* valt   = alloc(NBT);
  float* acc    = alloc(16);
  (void)ws_size; (void)n_in; (void)in_sizes; (void)out_size;

  k_init<<<1, 256, 0, stream>>>(zerov, zx2, acc);

  k_poincare<<<(NB * 32 + 255) / 256, 256, 0, stream>>>(lcas, lcas_h, NB);
  k_poincare<<<(NBT * 32 + 255) / 256, 256, 0, stream>>>(lcas_top, ltop_h, NBT);

  k_rownorm2<<<(NB + 7) / 8, 256, 0, stream>>>(lcas_h, y2, NB);
  k_rownorm2<<<(NBT + 7) / 8, 256, 0, stream>>>(ltop_h, y2t, NBT);
  k_rownorm2<<<BSZ / 8, 256, 0, stream>>>(z_s, x2, BSZ);
  k_rownorm2<<<BSZ / 8, 256, 0, stream>>>(z_s_1, x2_1, BSZ);

  // dominant WMMA GEMM: 1024 blocks x 8 waves, 16x16 tiles, K staged via TDM->LDS
  k_gemm_sim<<<BSZ / 16, 256, 2 * BUFSZ * sizeof(float), stream>>>(
      z_s, z_s_1, lcas_h, x2, x2_1, y2, sim, sim1);

  k_assign<<<1, 32, 0, stream>>>(sim, y, selp);
  k_row_losses<<<BSZ / 8, 256, 0, stream>>>(sim, sim1, selp, y, linear_w, acc);
  k_zdiff<<<1024, 256, 0, stream>>>(z_s, z_s_1, acc);

  k_dist_small<<<(NB * NB + 255) / 256, 256, 0, stream>>>(lcas_h, y2, NB, lcas_h, y2, NB, Dll);
  k_dist_small<<<(NB * NBT + 255) / 256, 256, 0, stream>>>(lcas_h, y2, NB, ltop_h, y2t, NBT, Dlt);
  k_dist_small<<<(NBT * NBT + 255) / 256, 256, 0, stream>>>(ltop_h, y2t, NBT, ltop_h, y2t, NBT, Dtt);
  k_dist_small<<<(NB + 255) / 256, 256, 0, stream>>>(zerov, zx2, 1, lcas_h, y2, NB, Dor);

  k_proto_lse<<<NB, 32, 0, stream>>>(Dll, NB, &acc[10]);
  k_proto_lse<<<1, 32, 0, stream>>>(Dor, NB, &acc[11]);

  k_build_simleaf<<<(NB * NB + 255) / 256, 256, 0, stream>>>(Dll, SL);
  k_build_simtop<<<(NBT * NBT + 255) / 256, 256, 0, stream>>>(Dtt, ST);

  k_topk_nn<<<NB, 32, 0, stream>>>(SL, NB, NNl);
  k_topk_nn<<<NBT, 32, 0, stream>>>(ST, NBT, NNt);

  k_triplets<<<(NB + 63) / 64, 64, 0, stream>>>(NNl, NB, ancl, posl, negl, vall, 0x1111u);
  k_triplets<<<(NBT + 63) / 64, 64, 0, stream>>>(NNt, NBT, anct, post, negt, valt, 0x2222u);

  k_ghhc<<<(NB * TPER + 63) / 64, 64, 0, stream>>>(Dlt, ancl, posl, negl, vall, NB * TPER,
                                                   0x3333u, &acc[6], &acc[7]);
  k_ghhc<<<(NBT * TPER + 63) / 64, 64, 0, stream>>>(Dtt, anct, post, negt, valt, NBT * TPER,
                                                    0x4444u, &acc[8], &acc[9]);

  k_finalize<<<1, 32, 0, stream>>>(acc, out);
}